// TalkingHeadAttention_72722386256423
// MI455X (gfx1250) — compile-verified
//
#include <hip/hip_runtime.h>
#include <hip/hip_bf16.h>

// Talking-heads attention, fp32, MI455X (gfx1250, wave32).
// Memory-bound (~100MB attn output mandatory) -> fp32 throughout, use
// V_WMMA_F32_16X16X4_F32 for every GEMM, fuse the 12x12 head mixes,
// async global->LDS staging + prefetch for the streamed panels.

typedef __attribute__((ext_vector_type(2))) float v2f;
typedef __attribute__((ext_vector_type(4))) float v4f;
typedef __attribute__((ext_vector_type(8))) float v8f;
typedef int v4i __attribute__((vector_size(16))); // matches async-LDS builtin param

#define BB 2
#define NN 1024
#define CC 768
#define HH 12
#define DHD 64
#define C3 2304
#define QKSCALE 0.125f // 64^-0.5

#define AS1 __attribute__((address_space(1)))
#define AS3 __attribute__((address_space(3)))

#if defined(__has_builtin)
#if __has_builtin(__builtin_amdgcn_global_load_async_to_lds_b128)
#define HAVE_ASYNC_COPY 1
#endif
#if __has_builtin(__builtin_amdgcn_s_wait_asynccnt)
#define HAVE_WAIT_ASYNC 1
#endif
#endif
#ifndef HAVE_ASYNC_COPY
#define HAVE_ASYNC_COPY 0
#endif
#ifndef HAVE_WAIT_ASYNC
#define HAVE_WAIT_ASYNC 0
#endif

// One 16-byte global->LDS copy. Async (ASYNCcnt) when available, else
// VGPR round-trip (covered by the barrier's dscnt wait).
__device__ __forceinline__ void copy16(const float* g, float* l) {
#if HAVE_ASYNC_COPY
  __builtin_amdgcn_global_load_async_to_lds_b128(
      (AS1 v4i*)g, (AS3 v4i*)l, 0, 0);
#else
  *(v4f*)l = *(const v4f*)g;
#endif
}

__device__ __forceinline__ void wait_async() {
#if HAVE_ASYNC_COPY
#if HAVE_WAIT_ASYNC
  __builtin_amdgcn_s_wait_asynccnt(0);
#else
  asm volatile("s_wait_asynccnt 0x0" ::: "memory");
#endif
#endif
}

__device__ __forceinline__ v8f wmma4(v2f a, v2f b, v8f c) {
  // D = A(16x4,f32) * B(4x16,f32) + C(16x16,f32)
  return __builtin_amdgcn_wmma_f32_16x16x4_f32(
      /*neg_a=*/false, a, /*neg_b=*/false, b,
      /*c_mod=*/(short)0, c, /*reuse_a=*/false, /*reuse_b=*/false);
}

// Inner product over a K=64 chunk for one 16x16 tile.
// sA  : 16 rows x 64 (row-major) -> A fragment: lane16 = M, kh2 selects K pair
// sBt : [col][64] (B stored K-contiguous per output column)
__device__ __forceinline__ v8f mma_chunk64(const float* sA, const float* sBt,
                                           int aRow, int bCol, int kh2, v8f acc) {
#pragma unroll
  for (int kk = 0; kk < 64; kk += 4) {
    v2f a = *(const v2f*)&sA[aRow * 64 + kk + kh2];
    v2f b = *(const v2f*)&sBt[bCol * 64 + kk + kh2];
    acc = wmma4(a, b, acc);
  }
  return acc;
}

// ---------------------------------------------------------------------------
// Kernel 1: qkv = x @ w_qkv + b_qkv, scattered to q/k/v (B,H,N,DH), q scaled.
// Grid (128, 18): 16-row M tiles x 128-col blocks. Block 256 = 8 waves.
// ---------------------------------------------------------------------------
__global__ void __launch_bounds__(256)
qkv_kernel(const float* __restrict__ x, const float* __restrict__ w_qkv,
           const float* __restrict__ b_qkv,
           float* __restrict__ q, float* __restrict__ k, float* __restrict__ v) {
  __shared__ __align__(16) float sA[16 * 64];
  __shared__ __align__(16) float sBt[128 * 64];
  const int t = threadIdx.x, lid = t & 31, wave = t >> 5;
  const int lane16 = lid & 15, kh2 = (lid >> 4) * 2, rhalf = (lid >> 4) * 8;
  const int rbase = blockIdx.x * 16;
  const int cbase = blockIdx.y * 128;
  v8f acc = {0, 0, 0, 0, 0, 0, 0, 0};
  for (int kc = 0; kc < CC; kc += 64) {
    { // A tile: x[rbase..+15][kc..+63] (direct copy -> async)
      int row = t >> 4, col4 = (t & 15) * 4;
      copy16(&x[(rbase + row) * CC + kc + col4], &sA[row * 64 + col4]);
    }
#pragma unroll
    for (int j = 0; j < 8; ++j) { // B tile transposed: sBt[col][krow]
      int e = t + j * 256, row = e >> 5, col4 = (e & 31) * 4;
      v4f tmp = *(const v4f*)&w_qkv[(kc + row) * C3 + cbase + col4];
      sBt[(col4 + 0) * 64 + row] = tmp.x;
      sBt[(col4 + 1) * 64 + row] = tmp.y;
      sBt[(col4 + 2) * 64 + row] = tmp.z;
      sBt[(col4 + 3) * 64 + row] = tmp.w;
    }
    if (kc + 64 < CC) // prefetch next B panel while this chunk computes
      __builtin_prefetch(&w_qkv[(kc + 64 + (t >> 5)) * C3 + cbase + (t & 31) * 4], 0, 0);
    wait_async();
    __syncthreads();
    acc = mma_chunk64(sA, sBt, lane16, wave * 16 + lane16, kh2, acc);
    __syncthreads();
  }
  const int c3 = cbase + wave * 16 + lane16;
  const int which = c3 / CC;             // 0=q 1=k 2=v
  const int c = c3 % CC, h = c / DHD, d = c % DHD;
  const float bias = b_qkv[c3];
  float* dst = (which == 0) ? q : (which == 1) ? k : v;
  const float sc = (which == 0) ? QKSCALE : 1.0f;
#pragma unroll
  for (int i = 0; i < 8; ++i) {
    int r = rbase + i + rhalf;
    int b_ = r >> 10, n = r & 1023;
    dst[((b_ * HH + h) * NN + n) * DHD + d] = (acc[i] + bias) * sc;
  }
}

// ---------------------------------------------------------------------------
// Kernel 2: logits[b][g][n][m] = sum_h w_l[h][g] * (q_h . k_h) + b_l[g]
// Grid (64, 8, 2): 16-n tile x 128-m block x batch. All 12 heads accumulated
// in registers (12 x v8f per wave), 12x12 mix applied in-register.
// ---------------------------------------------------------------------------
__global__ void __launch_bounds__(256)
logits_kernel(const float* __restrict__ q, const float* __restrict__ k,
              const float* __restrict__ w_l, const float* __restrict__ b_l,
              float* __restrict__ logits) {
  __shared__ __align__(16) float sQ[16 * 64];
  __shared__ __align__(16) float sK[128 * 64]; // [m][d] == B transposed already
  __shared__ float sWl[HH * HH];
  __shared__ float sBl[HH];
  const int t = threadIdx.x, lid = t & 31, wave = t >> 5;
  const int lane16 = lid & 15, kh2 = (lid >> 4) * 2, rhalf = (lid >> 4) * 8;
  const int nbase = blockIdx.x * 16;
  const int mbase = blockIdx.y * 128;
  const int b_ = blockIdx.z;
  if (t < HH * HH) sWl[t] = w_l[t];
  if (t < HH) sBl[t] = b_l[t];
  const v8f zero = {0, 0, 0, 0, 0, 0, 0, 0};
  v8f acc[HH];
#pragma unroll
  for (int h = 0; h < HH; ++h) acc[h] = zero;
#pragma unroll
  for (int h = 0; h < HH; ++h) {
    { int row = t >> 4, col4 = (t & 15) * 4;
      copy16(&q[((b_ * HH + h) * NN + nbase + row) * DHD + col4],
             &sQ[row * 64 + col4]); }
#pragma unroll
    for (int j = 0; j < 8; ++j) {
      int e = t + j * 256, row = e >> 4, col4 = (e & 15) * 4;
      copy16(&k[((b_ * HH + h) * NN + mbase + row) * DHD + col4],
             &sK[row * 64 + col4]);
    }
    if (h + 1 < HH) // prefetch next head's K panel (128 rows x 256B)
      __builtin_prefetch(
          &k[((b_ * HH + h + 1) * NN + mbase + (t >> 1)) * DHD + (t & 1) * 32], 0, 0);
    wait_async();
    __syncthreads();
    acc[h] = mma_chunk64(sQ, sK, lane16, wave * 16 + lane16, kh2, acc[h]);
    __syncthreads();
  }
  const int mcol = mbase + wave * 16 + lane16;
#pragma unroll
  for (int g = 0; g < HH; ++g) {
    v8f s;
#pragma unroll
    for (int i = 0; i < 8; ++i) s[i] = sBl[g];
#pragma unroll
    for (int h = 0; h < HH; ++h) {
      float w = sWl[h * HH + g];
#pragma unroll
      for (int i = 0; i < 8; ++i) s[i] += w * acc[h][i];
    }
#pragma unroll
    for (int i = 0; i < 8; ++i) {
      int n = nbase + i + rhalf;
      logits[((b_ * HH + g) * NN + n) * NN + mcol] = s[i];
    }
  }
}

// ---------------------------------------------------------------------------
// Kernel 3: per (b,n): softmax over m for each of 12 heads, then
// attn[g] = sum_h w_w[h][g]*softmax_h + b_w[g]. One workgroup per (b,n);
// 12x1024 floats (48KB) in LDS, wave32 shuffle reductions.
// ---------------------------------------------------------------------------
__global__ void __launch_bounds__(256)
softmax_mix2_kernel(const float* __restrict__ logits, const float* __restrict__ w_w,
                    const float* __restrict__ b_w, float* __restrict__ attn) {
  __shared__ __align__(16) float rows[HH][NN]; // 48 KB
  __shared__ float sWw[HH * HH], sBw[HH];
  __shared__ float sMax[HH], sInv[HH];
  const int t = threadIdx.x, lid = t & 31, wave = t >> 5;
  const int b_ = blockIdx.x >> 10, n = blockIdx.x & 1023;
  if (t < HH * HH) sWw[t] = w_w[t];
  if (t < HH) sBw[t] = b_w[t];
#pragma unroll
  for (int j = 0; j < HH; ++j) {
    int e = t + j * 256, h = e >> 8, m4 = (e & 255) * 4;
    copy16(&logits[((b_ * HH + h) * NN + n) * NN + m4], &rows[h][m4]);
  }
  wait_async();
  __syncthreads();
  for (int h = wave; h < HH; h += 8) { // waves 0..7 own heads {w, w+8}
    float mx = -3.402823466e38f;
    for (int m = lid; m < NN; m += 32) mx = fmaxf(mx, rows[h][m]);
#pragma unroll
    for (int off = 16; off > 0; off >>= 1) mx = fmaxf(mx, __shfl_xor(mx, off, 32));
    float sum = 0.f;
    for (int m = lid; m < NN; m += 32) sum += __expf(rows[h][m] - mx);
#pragma unroll
    for (int off = 16; off > 0; off >>= 1) sum += __shfl_xor(sum, off, 32);
    if (lid == 0) { sMax[h] = mx; sInv[h] = 1.0f / sum; }
  }
  __syncthreads();
  for (int m = t; m < NN; m += 256) {
    float p[HH];
#pragma unroll
    for (int h = 0; h < HH; ++h) p[h] = __expf(rows[h][m] - sMax[h]) * sInv[h];
#pragma unroll
    for (int g = 0; g < HH; ++g) {
      float o = sBw[g];
#pragma unroll
      for (int h = 0; h < HH; ++h) o += sWw[h * HH + g] * p[h];
      attn[((b_ * HH + g) * NN + n) * NN + m] = o;
    }
  }
}

// ---------------------------------------------------------------------------
// Kernel 4: oh[b][h][n][d] = sum_m attn[b][h][n][m] * v[b][h][m][d]
// Grid (32, 12, 2): 32-n tiles per (b,h). 8 waves = 2(n) x 4(d) 16x16 tiles.
// ---------------------------------------------------------------------------
__global__ void __launch_bounds__(256)
av_kernel(const float* __restrict__ attn, const float* __restrict__ v,
          float* __restrict__ oh) {
  __shared__ __align__(16) float sA[32 * 64];
  __shared__ __align__(16) float sBt[64 * 64]; // [d][m]
  const int t = threadIdx.x, lid = t & 31, wave = t >> 5;
  const int lane16 = lid & 15, kh2 = (lid >> 4) * 2, rhalf = (lid >> 4) * 8;
  const int rt = wave >> 2, ct = wave & 3;
  const int nbase = blockIdx.x * 32;
  const int h = blockIdx.y, b_ = blockIdx.z;
  const int baseA = (b_ * HH + h) * NN;
  v8f acc = {0, 0, 0, 0, 0, 0, 0, 0};
  for (int mc = 0; mc < NN; mc += 64) {
#pragma unroll
    for (int j = 0; j < 2; ++j) { // attn[nbase..+31][mc..+63] (direct -> async)
      int e = t + j * 256, row = e >> 4, col4 = (e & 15) * 4;
      copy16(&attn[(baseA + nbase + row) * NN + mc + col4], &sA[row * 64 + col4]);
    }
#pragma unroll
    for (int j = 0; j < 4; ++j) { // sBt[d][m] = v[mc+m][d] (transpose)
      int e = t + j * 256, row = e >> 4, col4 = (e & 15) * 4;
      v4f tmp = *(const v4f*)&v[(baseA + mc + row) * DHD + col4];
      sBt[(col4 + 0) * 64 + row] = tmp.x;
      sBt[(col4 + 1) * 64 + row] = tmp.y;
      sBt[(col4 + 2) * 64 + row] = tmp.z;
      sBt[(col4 + 3) * 64 + row] = tmp.w;
    }
    if (mc + 64 < NN) // prefetch next attn chunk (32 rows x 256B)
      __builtin_prefetch(
          &attn[(baseA + nbase + (t >> 3)) * NN + mc + 64 + (t & 7) * 8], 0, 0);
    wait_async();
    __syncthreads();
    acc = mma_chunk64(&sA[rt * 16 * 64], sBt, lane16, ct * 16 + lane16, kh2, acc);
    __syncthreads();
  }
#pragma unroll
  for (int i = 0; i < 8; ++i) {
    int nrow = nbase + rt * 16 + i + rhalf;
    oh[(baseA + nrow) * DHD + ct * 16 + lane16] = acc[i];
  }
}

// ---------------------------------------------------------------------------
// Kernel 5: out = reshape(oh) @ w_proj + b_proj. K-chunks of 64 align with
// exactly one head's contiguous DH block of oh.
// ---------------------------------------------------------------------------
__global__ void __launch_bounds__(256)
proj_kernel(const float* __restrict__ oh, const float* __restrict__ w_proj,
            const float* __restrict__ b_proj, float* __restrict__ out) {
  __shared__ __align__(16) float sA[16 * 64];
  __shared__ __align__(16) float sBt[128 * 64];
  const int t = threadIdx.x, lid = t & 31, wave = t >> 5;
  const int lane16 = lid & 15, kh2 = (lid >> 4) * 2, rhalf = (lid >> 4) * 8;
  const int rbase = blockIdx.x * 16;
  const int cbase = blockIdx.y * 128;
  const int b_ = rbase >> 10, n0 = rbase & 1023;
  v8f acc = {0, 0, 0, 0, 0, 0, 0, 0};
  for (int kc = 0; kc < CC; kc += 64) {
    const int h = kc >> 6; // this K-chunk == head h's DH block
    { int row = t >> 4, col4 = (t & 15) * 4;
      copy16(&oh[((b_ * HH + h) * NN + n0 + row) * DHD + col4],
             &sA[row * 64 + col4]); }
#pragma unroll
    for (int j = 0; j < 8; ++j) {
      int e = t + j * 256, row = e >> 5, col4 = (e & 31) * 4;
      v4f tmp = *(const v4f*)&w_proj[(kc + row) * CC + cbase + col4];
      sBt[(col4 + 0) * 64 + row] = tmp.x;
      sBt[(col4 + 1) * 64 + row] = tmp.y;
      sBt[(col4 + 2) * 64 + row] = tmp.z;
      sBt[(col4 + 3) * 64 + row] = tmp.w;
    }
    if (kc + 64 < CC)
      __builtin_prefetch(&w_proj[(kc + 64 + (t >> 5)) * CC + cbase + (t & 31) * 4], 0, 0);
    wait_async();
    __syncthreads();
    acc = mma_chunk64(sA, sBt, lane16, wave * 16 + lane16, kh2, acc);
    __syncthreads();
  }
  const int col = cbase + wave * 16 + lane16;
  const float bias = b_proj[col];
#pragma unroll
  for (int i = 0; i < 8; ++i) {
    int r = rbase + i + rhalf;
    out[r * CC + col] = acc[i] + bias;
  }
}

// ---------------------------------------------------------------------------
extern "C" void kernel_launch(void* const* d_in, const int* in_sizes, int n_in,
                              void* d_out, int out_size, void* d_ws, size_t ws_size,
                              hipStream_t stream) {
  const float* x      = (const float*)d_in[0];
  const float* w_qkv  = (const float*)d_in[1];
  const float* b_qkv  = (const float*)d_in[2];
  const float* w_l    = (const float*)d_in[3];
  const float* b_l    = (const float*)d_in[4];
  const float* w_w    = (const float*)d_in[5];
  const float* b_w    = (const float*)d_in[6];
  const float* w_proj = (const float*)d_in[7];
  const float* b_proj = (const float*)d_in[8];

  float* out  = (float*)d_out;                         // (B,N,C)
  float* attn = out + (size_t)BB * NN * CC;            // (B,H,N,N), also output

  float* ws     = (float*)d_ws;
  float* q      = ws;                                  // (B,H,N,DH)
  float* k      = q + (size_t)BB * HH * NN * DHD;
  float* v      = k + (size_t)BB * HH * NN * DHD;
  float* logits = v + (size_t)BB * HH * NN * DHD;      // (B,H,N,N) pre-softmax (mixed)
  float* oh     = logits + (size_t)BB * HH * NN * NN;  // (B,H,N,DH)

  qkv_kernel<<<dim3(128, 18), 256, 0, stream>>>(x, w_qkv, b_qkv, q, k, v);
  logits_kernel<<<dim3(64, 8, 2), 256, 0, stream>>>(q, k, w_l, b_l, logits);
  softmax_mix2_kernel<<<dim3(2048), 256, 0, stream>>>(logits, w_w, b_w, attn);
  av_kernel<<<dim3(32, 12, 2), 256, 0, stream>>>(attn, v, oh);
  proj_kernel<<<dim3(128, 6), 256, 0, stream>>>(oh, w_proj, b_proj, out);
}